// NCFBinary2_41875931136474
// MI455X (gfx1250) — compile-verified
//
#include <hip/hip_runtime.h>
#include <hip/hip_bf16.h>
#include <math.h>

typedef __attribute__((ext_vector_type(2))) float v2f;
typedef __attribute__((ext_vector_type(8))) float v8f;

#define WAVES 8
#define BLOCK 256
#define E_DIM 16
#define F_DIM 19
#define H_DIM 128
#define D_DIM 64
#define X_DIM 48
#define NEXP 3
#define EPS 1e-5f

// LDS budget (~213 KB, under 320 KB/WGP):
//  wsh_t   : [128][48]        24576 B  (W_sh transposed: [n][k])
//  wexp_t  : [3][64][128]     98304 B  (W_exp transposed: [i][n][k])
//  x stage : [8][16][48]      24576 B
//  s stage : [8][16][128]     65536 B  (reused for expert h[16][64])
//  misc    : ~4.8 KB

__global__ __launch_bounds__(BLOCK)
void NCFBinary2_41875931136474_kernel(
    const int*   __restrict__ home_team,
    const int*   __restrict__ away_team,
    const float* __restrict__ game_features,
    const int*   __restrict__ sport_id,
    const float* __restrict__ emb_home,
    const float* __restrict__ emb_away,
    const float* __restrict__ W_gf,
    const float* __restrict__ b_gf,
    const float* __restrict__ W_sh,
    const float* __restrict__ b_sh,
    const float* __restrict__ bn_gamma,
    const float* __restrict__ bn_beta,
    const float* __restrict__ bn_mean,
    const float* __restrict__ bn_var,
    const float* __restrict__ W_exp,
    const float* __restrict__ b_exp,
    const float* __restrict__ ln_gamma,
    const float* __restrict__ ln_beta,
    const float* __restrict__ W_out,
    const float* __restrict__ b_out,
    float*       __restrict__ out,
    int B)
{
    __shared__ __align__(16) float lds_wsh[H_DIM * X_DIM];          // [n][k]
    __shared__ __align__(16) float lds_wexp[NEXP * D_DIM * H_DIM];  // [i][n][k]
    __shared__ __align__(16) float lds_x[WAVES * 16 * X_DIM];       // per-wave x tile
    __shared__ __align__(16) float lds_s[WAVES * 16 * H_DIM];       // per-wave s tile / h tile
    __shared__ float lds_scale[H_DIM], lds_bias[H_DIM];
    __shared__ float lds_bexp[NEXP * D_DIM], lds_lng[NEXP * D_DIM], lds_lnb[NEXP * D_DIM];
    __shared__ float lds_wgf[F_DIM * E_DIM], lds_bgf[E_DIM], lds_wout[D_DIM];

    const int tid  = threadIdx.x;
    const int wave = tid >> 5;
    const int lane = tid & 31;
    const int half = lane >> 4;        // 0: K_local 0,1 / M r ; 1: K_local 2,3 / M r+8
    const int row  = lane & 15;        // matrix row (A) / column-in-tile (B,C)

    // ---- Stage 0: cooperative weight staging into LDS -------------------
    for (int idx = tid; idx < H_DIM * X_DIM; idx += BLOCK) {
        int n = idx / X_DIM, k = idx % X_DIM;
        lds_wsh[idx] = W_sh[k * H_DIM + n];
    }
    for (int idx = tid; idx < NEXP * D_DIM * H_DIM; idx += BLOCK) {
        int i = idx / (D_DIM * H_DIM);
        int r = idx % (D_DIM * H_DIM);
        int n = r / H_DIM, k = r % H_DIM;
        lds_wexp[idx] = W_exp[(i * H_DIM + k) * D_DIM + n];
    }
    if (tid < H_DIM) {
        float sc = bn_gamma[tid] * rsqrtf(bn_var[tid] + EPS);
        lds_scale[tid] = sc;
        lds_bias[tid]  = (b_sh[tid] - bn_mean[tid]) * sc + bn_beta[tid];
    }
    for (int idx = tid; idx < NEXP * D_DIM; idx += BLOCK) {
        lds_bexp[idx] = b_exp[idx];
        lds_lng[idx]  = ln_gamma[idx];
        lds_lnb[idx]  = ln_beta[idx];
    }
    for (int idx = tid; idx < F_DIM * E_DIM; idx += BLOCK) lds_wgf[idx] = W_gf[idx];
    if (tid < E_DIM) lds_bgf[tid] = b_gf[tid];
    if (tid < D_DIM) lds_wout[tid] = W_out[tid];   // W_out is [64,1]
    __syncthreads();

    const float bo = b_out[0];
    const int nTiles = B >> 4;   // B assumed multiple of 16 (262144)

    float* xw = &lds_x[wave * 16 * X_DIM];
    float* sw = &lds_s[wave * 16 * H_DIM];

    for (int tile = blockIdx.x * WAVES + wave; tile < nTiles;
         tile += gridDim.x * WAVES) {

        // ---- Stage A: build x[16][48] = [he | ae | relu(gf@Wgf+bgf)] ----
        const int sample = tile * 16 + row;            // 2 lanes per sample
        const int home   = home_team[sample];
        const int away   = away_team[sample];
        const int sport  = sport_id[sample];

        {
            const float* hp = emb_home + (size_t)home * E_DIM + half * 8;
            const float* ap = emb_away + (size_t)away * E_DIM + half * 8;
            float* xr = &xw[row * X_DIM];
            #pragma unroll
            for (int j = 0; j < 8; ++j) xr[half * 8 + j] = hp[j];
            #pragma unroll
            for (int j = 0; j < 8; ++j) xr[E_DIM + half * 8 + j] = ap[j];

            const float* gp = game_features + (size_t)sample * F_DIM;
            float feats[F_DIM];
            #pragma unroll
            for (int f = 0; f < F_DIM; ++f) feats[f] = gp[f];
            #pragma unroll
            for (int j = 0; j < 8; ++j) {
                int e = half * 8 + j;
                float acc = lds_bgf[e];
                #pragma unroll
                for (int f = 0; f < F_DIM; ++f)
                    acc = fmaf(feats[f], lds_wgf[f * E_DIM + e], acc);
                xr[2 * E_DIM + e] = fmaxf(acc, 0.0f);
            }
        }

        // ---- Stage B: s = relu(affine(x @ W_sh))  [16x48]@[48x128] ------
        // A-fragments: K_local = vgpr + 2*half, M = row
        v2f afr[12];
        #pragma unroll
        for (int k = 0; k < 12; ++k)
            afr[k] = *(const v2f*)&xw[row * X_DIM + 4 * k + 2 * half];

        #pragma unroll
        for (int n = 0; n < 8; ++n) {
            v8f c = {};
            #pragma unroll
            for (int k = 0; k < 12; ++k) {
                v2f b = *(const v2f*)&lds_wsh[(n * 16 + row) * X_DIM + 4 * k + 2 * half];
                c = __builtin_amdgcn_wmma_f32_16x16x4_f32(
                        false, afr[k], false, b, (short)0, c, false, false);
            }
            const int col = n * 16 + row;               // per-lane column, same for all r
            const float sc = lds_scale[col], bi = lds_bias[col];
            #pragma unroll
            for (int r = 0; r < 8; ++r) {
                float v = fmaxf(fmaf(c[r], sc, bi), 0.0f);
                sw[(r + 8 * half) * H_DIM + col] = v;
            }
        }

        // Hoist s A-fragments into registers (frees the s staging buffer)
        v2f sfr[32];
        #pragma unroll
        for (int k = 0; k < 32; ++k)
            sfr[k] = *(const v2f*)&sw[row * H_DIM + 4 * k + 2 * half];

        // ---- Stage C: experts h = s @ W_exp[i] + b, LN, relu, dot, sigmoid
        for (int i = 0; i < NEXP; ++i) {
            #pragma unroll
            for (int n = 0; n < 4; ++n) {
                v8f c = {};
                #pragma unroll
                for (int k = 0; k < 32; ++k) {
                    v2f b = *(const v2f*)&lds_wexp[((i * D_DIM + n * 16 + row) * H_DIM)
                                                   + 4 * k + 2 * half];
                    c = __builtin_amdgcn_wmma_f32_16x16x4_f32(
                            false, sfr[k], false, b, (short)0, c, false, false);
                }
                const int col = n * 16 + row;
                const float be = lds_bexp[i * D_DIM + col];
                #pragma unroll
                for (int r = 0; r < 8; ++r)
                    sw[(r + 8 * half) * H_DIM + col] = c[r] + be;   // h tile (cols 0..63)
            }

            // LayerNorm over D=64 per row: lane covers 32 cols, pair via shfl_xor(16)
            float hv[32];
            const int base = row * H_DIM + half * 32;
            #pragma unroll
            for (int j = 0; j < 32; ++j) hv[j] = sw[base + j];

            float sum = 0.f, ss = 0.f;
            #pragma unroll
            for (int j = 0; j < 32; ++j) { sum += hv[j]; ss = fmaf(hv[j], hv[j], ss); }
            sum += __shfl_xor(sum, 16, 32);
            ss  += __shfl_xor(ss, 16, 32);
            const float mean = sum * (1.0f / 64.0f);
            const float var  = ss * (1.0f / 64.0f) - mean * mean;
            const float inv  = rsqrtf(var + EPS);

            if (sport == i) {   // partner lanes share sample -> pair stays convergent
                float acc = 0.f;
                #pragma unroll
                for (int j = 0; j < 32; ++j) {
                    int c2 = half * 32 + j;
                    float hn = fmaf((hv[j] - mean) * inv, lds_lng[i * D_DIM + c2],
                                    lds_lnb[i * D_DIM + c2]);
                    hn = fmaxf(hn, 0.0f);
                    acc = fmaf(hn, lds_wout[c2], acc);
                }
                acc += __shfl_xor(acc, 16, 32);
                if (half == 0)
                    out[sample] = 1.0f / (1.0f + expf(-(acc + bo)));
            }
        }
    }
}

extern "C" void kernel_launch(void* const* d_in, const int* in_sizes, int n_in,
                              void* d_out, int out_size, void* d_ws, size_t ws_size,
                              hipStream_t stream) {
    const int*   home_team     = (const int*)  d_in[0];
    const int*   away_team     = (const int*)  d_in[1];
    const float* game_features = (const float*)d_in[2];
    const int*   sport_id      = (const int*)  d_in[3];
    const float* emb_home      = (const float*)d_in[4];
    const float* emb_away      = (const float*)d_in[5];
    const float* W_gf          = (const float*)d_in[6];
    const float* b_gf          = (const float*)d_in[7];
    const float* W_sh          = (const float*)d_in[8];
    const float* b_sh          = (const float*)d_in[9];
    const float* bn_gamma      = (const float*)d_in[10];
    const float* bn_beta       = (const float*)d_in[11];
    const float* bn_mean       = (const float*)d_in[12];
    const float* bn_var        = (const float*)d_in[13];
    const float* W_exp         = (const float*)d_in[14];
    const float* b_exp         = (const float*)d_in[15];
    const float* ln_gamma      = (const float*)d_in[16];
    const float* ln_beta       = (const float*)d_in[17];
    const float* W_out         = (const float*)d_in[18];
    const float* b_out         = (const float*)d_in[19];

    const int B = in_sizes[0];
    const int nTiles = B / 16;
    int blocks = (nTiles + WAVES - 1) / WAVES;
    if (blocks > 512) blocks = 512;   // grid-stride; amortize LDS weight staging

    NCFBinary2_41875931136474_kernel<<<blocks, BLOCK, 0, stream>>>(
        home_team, away_team, game_features, sport_id,
        emb_home, emb_away, W_gf, b_gf, W_sh, b_sh,
        bn_gamma, bn_beta, bn_mean, bn_var,
        W_exp, b_exp, ln_gamma, ln_beta, W_out, b_out,
        (float*)d_out, B);
}